// VoxCPM2DecodeStepWrapper_65712999629512
// MI455X (gfx1250) — compile-verified
//
#include <hip/hip_runtime.h>
#include <math.h>

// MiniCPM decode-step: L=8, B=4, H=16, HKV=8, D=64, S=2048, HID=1024, FF=2816
#define L_    8
#define B_    4
#define H_    16
#define HKV_  8
#define D_    64
#define S_    2048
#define HID_  1024
#define FF_   2816
#define EPS_  1e-5f
#define MUP_  0.49497474683058327f   // 1.4 / sqrt(8)
#define ISQD_ 0.125f                 // 1 / sqrt(64)

typedef __attribute__((ext_vector_type(2))) float v2f;
typedef __attribute__((ext_vector_type(8))) float v8f;

// ---- workspace layout (float offsets) ----
#define WS_H    0        // [B,HID]       4096
#define WS_HN   4096     // [B,HID]       4096
#define WS_Q    8192     // [B,H*D]       4096
#define WS_K    12288    // [B,HKV*D]     2048
#define WS_V    14336    // [B,HKV*D]     2048
#define WS_O    16384    // [B,H*D]       4096
#define WS_ACT  20480    // [B,FF]        11264

#define WMMA_F32(a, b, c) \
  __builtin_amdgcn_wmma_f32_16x16x4_f32(false, (a), false, (b), (short)0, (c), false, false)

// ============================================================================
// Stage contiguous A[4][K] into LDS and append one zero row (row 4) so the
// M-padding lanes (m >= 4) read zeros with NO exec-masked loads.
// Block = 256 threads.
// ============================================================================
template <int K>
__device__ __forceinline__ void stage_A(const float* __restrict__ A, float* As) {
  float4* dst = (float4*)As;
  const float4* src = (const float4*)A;
  for (int i = threadIdx.x; i < K; i += 256) dst[i] = src[i];        // 4*K floats
  const float4 z = {0.f, 0.f, 0.f, 0.f};
  float4* zr = (float4*)(As + 4 * K);
  for (int i = threadIdx.x; i < K / 4; i += 256) zr[i] = z;          // zero row
  __syncthreads();
}

// ============================================================================
// One wave computes a K/8 slice of a 16x16 fp32 WMMA tile (cols [n0,n0+16)),
// using TWO interleaved accumulator chains for load/WMMA overlap.
// A-operand from LDS (ds_load_b64), B-operand streamed from global (read once).
// ============================================================================
template <int K>
__device__ __forceinline__ v8f wmma_slice2(const float* As,
                                           const float* __restrict__ W,
                                           int ldW, int n0) {
  const int lane = threadIdx.x & 31;
  const int m    = lane & 15;
  const int kh   = (lane >> 4) & 1;            // K sub-pair {0,1} vs {2,3}
  const int mrow = (m < B_) ? m : B_;          // padded lanes -> zero row
  const float* arow = As + mrow * K;
  const int Kw   = K / 8;                      // per-wave K slice
  const int kbeg = (threadIdx.x >> 5) * Kw;
  const int half = Kw / 2;
  v8f c0 = {}, c1 = {};
  for (int k = 0; k < half; k += 4) {
    const int ka0 = kbeg + k + 2 * kh;
    const int ka1 = ka0 + half;
    v2f a0 = *(const v2f*)(arow + ka0);        // ds_load_b64
    v2f a1 = *(const v2f*)(arow + ka1);
    const float* w0 = W + (size_t)ka0 * ldW + n0 + m;
    const float* w1 = W + (size_t)ka1 * ldW + n0 + m;
    v2f b0, b1;
    b0.x = w0[0];   b0.y = w0[ldW];
    b1.x = w1[0];   b1.y = w1[ldW];
    c0 = WMMA_F32(a0, b0, c0);
    c1 = WMMA_F32(a1, b1, c1);
  }
  return c0 + c1;
}

// part: [8 waves][4 rows][16 cols]
__device__ __forceinline__ void store_partial(float* part, const v8f& c) {
  const int wave = threadIdx.x >> 5;
  const int lane = threadIdx.x & 31;
  if (lane < 16)
    for (int r = 0; r < B_; ++r) part[wave * 64 + r * 16 + lane] = c[r];
}

__device__ __forceinline__ float reduce8(const float* part) {
  float s = 0.f;                                // caller: threadIdx.x < 64
  for (int w = 0; w < 8; ++w) s += part[w * 64 + threadIdx.x];
  return s;
}

// ============================================================================
__global__ void k_copy(const float* __restrict__ x, float* __restrict__ h, int n) {
  int i = blockIdx.x * blockDim.x + threadIdx.x;
  if (i < n) h[i] = x[i];
}

__global__ void k_rmsnorm(const float* __restrict__ in, const float* __restrict__ w,
                          float* __restrict__ out) {
  __shared__ float red[256];
  const int b = blockIdx.x;
  const float* r = in + b * HID_;
  float ss = 0.0f;
  for (int i = threadIdx.x; i < HID_; i += 256) { float v = r[i]; ss += v * v; }
  red[threadIdx.x] = ss; __syncthreads();
  for (int t = 128; t > 0; t >>= 1) {
    if (threadIdx.x < t) red[threadIdx.x] += red[threadIdx.x + t];
    __syncthreads();
  }
  const float inv = rsqrtf(red[0] / (float)HID_ + EPS_);
  for (int i = threadIdx.x; i < HID_; i += 256)
    out[b * HID_ + i] = r[i] * inv * w[i];
}

// QKV: one block per 16-col tile (128 blocks); 8 waves K-split + LDS reduce.
__global__ void k_qkv(const float* __restrict__ hn,
                      const float* __restrict__ Wq, const float* __restrict__ Wk,
                      const float* __restrict__ Wv,
                      float* __restrict__ q, float* __restrict__ k,
                      float* __restrict__ v) {
  __shared__ float As[5 * HID_];
  __shared__ float part[8 * 64];
  stage_A<HID_>(hn, As);
  const int t = blockIdx.x;
  const float* W; float* out; int ldW, n0;
  if (t < 64)      { W = Wq; out = q; ldW = H_ * D_;   n0 = t * 16; }
  else if (t < 96) { W = Wk; out = k; ldW = HKV_ * D_; n0 = (t - 64) * 16; }
  else             { W = Wv; out = v; ldW = HKV_ * D_; n0 = (t - 96) * 16; }
  v8f c = wmma_slice2<HID_>(As, W, ldW, n0);
  store_partial(part, c);
  __syncthreads();
  if (threadIdx.x < 64) {
    const float s = reduce8(part);
    const int r = threadIdx.x >> 4, n = threadIdx.x & 15;
    out[r * ldW + n0 + n] = s;
  }
}

// RoPE on q,k (pairs d / d+32) + emit k_upd (rope'd) and v_upd (raw)
__global__ void k_rope_store(float* __restrict__ q, float* __restrict__ k,
                             const float* __restrict__ v,
                             float* __restrict__ k_upd, float* __restrict__ v_upd,
                             const int* __restrict__ plen) {
  const int i   = blockIdx.x * blockDim.x + threadIdx.x;
  const int pos = *plen;
  const int QP = B_ * H_ * 32;     // 2048 q pairs
  const int KP = B_ * HKV_ * 32;   // 1024 k pairs
  const int VN = B_ * HKV_ * D_;   // 2048 v elems
  if (i < QP) {
    const int d = i & 31, head = i >> 5;
    const float ang = (float)pos * __powf(10000.0f, -(float)(2 * d) / 64.0f);
    const float cs = __cosf(ang), sn = __sinf(ang);
    const float x1 = q[head * 64 + d], x2 = q[head * 64 + d + 32];
    q[head * 64 + d]      = x1 * cs - x2 * sn;
    q[head * 64 + d + 32] = x2 * cs + x1 * sn;
  } else if (i < QP + KP) {
    const int j = i - QP, d = j & 31, head = j >> 5;
    const float ang = (float)pos * __powf(10000.0f, -(float)(2 * d) / 64.0f);
    const float cs = __cosf(ang), sn = __sinf(ang);
    const float x1 = k[head * 64 + d], x2 = k[head * 64 + d + 32];
    const float r1 = x1 * cs - x2 * sn, r2 = x2 * cs + x1 * sn;
    k[head * 64 + d]          = r1;  k[head * 64 + d + 32]     = r2;
    k_upd[head * 64 + d]      = r1;  k_upd[head * 64 + d + 32] = r2;
  } else if (i < QP + KP + VN) {
    const int j = i - QP - KP;
    v_upd[j] = v[j];
  }
}

// Attention: one block per (batch, query head). KV cache is read-only; s==pos
// substitutes the freshly computed k/v (reference's where(write_mask,...)).
__global__ void k_attn(const float* __restrict__ q, const float* __restrict__ knew,
                       const float* __restrict__ vnew,
                       const float* __restrict__ kc, const float* __restrict__ vc,
                       float* __restrict__ o, const int* __restrict__ plen,
                       int layer) {
  __shared__ float sc[S_];
  __shared__ float red[256];
  __shared__ float oacc[4 * D_];
  __shared__ __align__(16) float qs[D_];
  const int b = blockIdx.x >> 4;
  const int h = blockIdx.x & 15;
  const int kvh = h >> 1;
  const int pos = *plen;
  const float* kbase = kc + ((((size_t)layer * B_ + b) * HKV_ + kvh) * S_) * D_;
  const float* vbase = vc + ((((size_t)layer * B_ + b) * HKV_ + kvh) * S_) * D_;
  const float* kn = knew + (b * HKV_ + kvh) * D_;
  const float* vn = vnew + (b * HKV_ + kvh) * D_;
  if (threadIdx.x < D_) qs[threadIdx.x] = q[(b * H_ + h) * D_ + threadIdx.x];
  __syncthreads();

  float mx = -1e30f;
  const float4* qs4 = (const float4*)qs;
  for (int s = threadIdx.x; s <= pos; s += 256) {
    const float* kr = (s == pos) ? kn : (kbase + (size_t)s * D_);
    const float4* kr4 = (const float4*)kr;
    float acc = 0.0f;
    for (int dd = 0; dd < D_ / 4; ++dd) {
      const float4 kk = kr4[dd], qq = qs4[dd];
      acc += qq.x * kk.x + qq.y * kk.y + qq.z * kk.z + qq.w * kk.w;
    }
    acc *= ISQD_;
    sc[s] = acc;
    mx = fmaxf(mx, acc);
  }
  red[threadIdx.x] = mx; __syncthreads();
  for (int t = 128; t > 0; t >>= 1) {
    if (threadIdx.x < t) red[threadIdx.x] = fmaxf(red[threadIdx.x], red[threadIdx.x + t]);
    __syncthreads();
  }
  mx = red[0]; __syncthreads();

  float sum = 0.0f;
  for (int s = threadIdx.x; s <= pos; s += 256) {
    const float e = __expf(sc[s] - mx);
    sc[s] = e; sum += e;
  }
  red[threadIdx.x] = sum; __syncthreads();
  for (int t = 128; t > 0; t >>= 1) {
    if (threadIdx.x < t) red[threadIdx.x] += red[threadIdx.x + t];
    __syncthreads();
  }
  const float inv = 1.0f / red[0]; __syncthreads();

  const int d = threadIdx.x & 63;
  const int g = threadIdx.x >> 6;          // 4 s-groups
  float acc = 0.0f;
  for (int s = g; s <= pos; s += 4) {
    const float* vr = (s == pos) ? vn : (vbase + (size_t)s * D_);
    acc += sc[s] * vr[d];
  }
  oacc[g * D_ + d] = acc; __syncthreads();
  if (threadIdx.x < D_) {
    const float t = oacc[threadIdx.x] + oacc[D_ + threadIdx.x] +
                    oacc[2 * D_ + threadIdx.x] + oacc[3 * D_ + threadIdx.x];
    o[(b * H_ + h) * D_ + threadIdx.x] = t * inv;
  }
}

// h += (o @ Wo) * mup     (64 blocks, K-split)
__global__ void k_oproj(const float* __restrict__ o, const float* __restrict__ Wo,
                        float* __restrict__ h) {
  __shared__ float As[5 * HID_];
  __shared__ float part[8 * 64];
  stage_A<HID_>(o, As);
  const int n0 = blockIdx.x * 16;
  v8f c = wmma_slice2<HID_>(As, Wo, HID_, n0);
  store_partial(part, c);
  __syncthreads();
  if (threadIdx.x < 64) {
    const float s = reduce8(part);
    const int r = threadIdx.x >> 4, n = threadIdx.x & 15;
    h[r * HID_ + n0 + n] += MUP_ * s;
  }
}

// act = silu(hn @ Wg) * (hn @ Wu)   (176 blocks; gate/up share the A load)
__global__ void k_gateup(const float* __restrict__ hn, const float* __restrict__ Wg,
                         const float* __restrict__ Wu, float* __restrict__ act) {
  __shared__ float As[5 * HID_];
  __shared__ float pg[8 * 64];
  __shared__ float pu[8 * 64];
  stage_A<HID_>(hn, As);
  const int n0 = blockIdx.x * 16;

  const int lane = threadIdx.x & 31;
  const int m    = lane & 15;
  const int kh   = (lane >> 4) & 1;
  const int mrow = (m < B_) ? m : B_;
  const float* arow = As + mrow * HID_;
  const int Kw   = HID_ / 8;
  const int kbeg = (threadIdx.x >> 5) * Kw;
  v8f cg = {}, cu = {};
  for (int k = 0; k < Kw; k += 4) {
    const int ka = kbeg + k + 2 * kh;
    v2f a = *(const v2f*)(arow + ka);
    const float* wg = Wg + (size_t)ka * FF_ + n0 + m;
    const float* wu = Wu + (size_t)ka * FF_ + n0 + m;
    v2f bg, bu;
    bg.x = wg[0];  bg.y = wg[FF_];
    bu.x = wu[0];  bu.y = wu[FF_];
    cg = WMMA_F32(a, bg, cg);
    cu = WMMA_F32(a, bu, cu);
  }
  store_partial(pg, cg);
  store_partial(pu, cu);
  __syncthreads();
  if (threadIdx.x < 64) {
    const float g = reduce8(pg);
    const float u = reduce8(pu);
    const int r = threadIdx.x >> 4, n = threadIdx.x & 15;
    act[r * FF_ + n0 + n] = (g / (1.0f + __expf(-g))) * u;
  }
}

// h += (act @ Wd) * mup    (64 blocks, K = 2816 split 8 ways)
__global__ void k_down(const float* __restrict__ act, const float* __restrict__ Wd,
                       float* __restrict__ h) {
  __shared__ float As[5 * FF_];     // 56.3 KB staging
  __shared__ float part[8 * 64];
  stage_A<FF_>(act, As);
  const int n0 = blockIdx.x * 16;
  v8f c = wmma_slice2<FF_>(As, Wd, HID_, n0);
  store_partial(part, c);
  __syncthreads();
  if (threadIdx.x < 64) {
    const float s = reduce8(part);
    const int r = threadIdx.x >> 4, n = threadIdx.x & 15;
    h[r * HID_ + n0 + n] += MUP_ * s;
  }
}

// final rmsnorm -> d_out[0:4096]; emit pos+1 (int bits) at tail
__global__ void k_final(const float* __restrict__ h, const float* __restrict__ w,
                        float* __restrict__ out, const int* __restrict__ plen,
                        int* __restrict__ posout) {
  __shared__ float red[256];
  const int b = blockIdx.x;
  const float* r = h + b * HID_;
  float ss = 0.0f;
  for (int i = threadIdx.x; i < HID_; i += 256) { float v = r[i]; ss += v * v; }
  red[threadIdx.x] = ss; __syncthreads();
  for (int t = 128; t > 0; t >>= 1) {
    if (threadIdx.x < t) red[threadIdx.x] += red[threadIdx.x + t];
    __syncthreads();
  }
  const float inv = rsqrtf(red[0] / (float)HID_ + EPS_);
  for (int i = threadIdx.x; i < HID_; i += 256)
    out[b * HID_ + i] = r[i] * inv * w[i];
  if (b == 0 && threadIdx.x == 0) *posout = *plen + 1;
}

// ============================================================================
extern "C" void kernel_launch(void* const* d_in, const int* in_sizes, int n_in,
                              void* d_out, int out_size, void* d_ws, size_t ws_size,
                              hipStream_t stream) {
  const float* x     = (const float*)d_in[0];
  const float* kc    = (const float*)d_in[1];
  const float* vc    = (const float*)d_in[2];
  const float* Wq    = (const float*)d_in[3];
  const float* Wk    = (const float*)d_in[4];
  const float* Wv    = (const float*)d_in[5];
  const float* Wo    = (const float*)d_in[6];
  const float* Wg    = (const float*)d_in[7];
  const float* Wu    = (const float*)d_in[8];
  const float* Wd    = (const float*)d_in[9];
  const float* n1w   = (const float*)d_in[10];
  const float* n2w   = (const float*)d_in[11];
  const float* fnw   = (const float*)d_in[12];
  const int*   plen  = (const int*)d_in[13];

  float* out  = (float*)d_out;
  float* ws   = (float*)d_ws;
  float* h    = ws + WS_H;
  float* hn   = ws + WS_HN;
  float* qb   = ws + WS_Q;
  float* kb   = ws + WS_K;
  float* vb   = ws + WS_V;
  float* ob   = ws + WS_O;
  float* act  = ws + WS_ACT;

  float* kupd   = out + B_ * HID_;                               // 4096
  float* vupd   = kupd + L_ * B_ * HKV_ * D_;                    // +16384
  int*   posout = (int*)(vupd + L_ * B_ * HKV_ * D_);            // +16384

  k_copy<<<16, 256, 0, stream>>>(x, h, B_ * HID_);

  for (int l = 0; l < L_; ++l) {
    const float* wq = Wq + (size_t)l * HID_ * H_ * D_;
    const float* wk = Wk + (size_t)l * HID_ * HKV_ * D_;
    const float* wv = Wv + (size_t)l * HID_ * HKV_ * D_;
    const float* wo = Wo + (size_t)l * H_ * D_ * HID_;
    const float* wg = Wg + (size_t)l * HID_ * FF_;
    const float* wu = Wu + (size_t)l * HID_ * FF_;
    const float* wd = Wd + (size_t)l * FF_ * HID_;

    k_rmsnorm<<<B_, 256, 0, stream>>>(h, n1w + l * HID_, hn);
    k_qkv<<<128, 256, 0, stream>>>(hn, wq, wk, wv, qb, kb, vb);
    k_rope_store<<<20, 256, 0, stream>>>(qb, kb, vb,
                                         kupd + l * B_ * HKV_ * D_,
                                         vupd + l * B_ * HKV_ * D_, plen);
    k_attn<<<B_ * H_, 256, 0, stream>>>(qb, kb, vb, kc, vc, ob, plen, l);
    k_oproj<<<64, 256, 0, stream>>>(ob, wo, h);
    k_rmsnorm<<<B_, 256, 0, stream>>>(h, n2w + l * HID_, hn);
    k_gateup<<<FF_ / 16, 256, 0, stream>>>(hn, wg, wu, act);
    k_down<<<64, 256, 0, stream>>>(act, wd, h);
  }

  k_final<<<B_, 256, 0, stream>>>(h, fnw, out, plen, posout);
}